// MOE_10488310137290
// MI455X (gfx1250) — compile-verified
//
#include <hip/hip_runtime.h>

typedef __attribute__((ext_vector_type(2))) float v2f;
typedef __attribute__((ext_vector_type(8))) float v8f;

#define TT 1024   // tokens (B*S)
#define HH 1024   // hidden
#define EE 8      // experts
#define II 1408   // intermediate
#define KC 32     // K chunk staged in LDS per double-buffer step
#define APAD 36   // A-tile row stride (floats): bank-conflict-free, 16B rows
#define BPAD 136  // B-tile row stride (floats): bank-conflict-free, 16B rows

static __device__ __forceinline__ v8f wmma_f32(v2f a, v2f b, v8f c) {
  // D = A(16x4 f32) * B(4x16 f32) + C(16x16 f32)
  return __builtin_amdgcn_wmma_f32_16x16x4_f32(false, a, false, b, (short)0, c,
                                               false, false);
}

// One 16-byte global -> LDS async DMA (ASYNCcnt-tracked, no VGPR staging).
static __device__ __forceinline__ void async_b128(unsigned lds_off,
                                                  const float* gsrc) {
  asm volatile("global_load_async_to_lds_b128 %0, %1, off"
               :
               : "v"(lds_off), "v"(gsrc)
               : "memory");
}
#define LDSOFF(p) ((unsigned)(size_t)(p))
#define WAIT_ASYNC(n) asm volatile("s_wait_asynccnt " #n ::: "memory")

// ---------------------------------------------------------------- zero output
__global__ __launch_bounds__(256) void moe_zero(float4* __restrict__ out) {
  out[blockIdx.x * 256 + threadIdx.x] = make_float4(0.f, 0.f, 0.f, 0.f);
}

// ---------------------------------------------------------------- router
// One wave (32 lanes) per token. grid = T/8 blocks of 256 threads.
__global__ __launch_bounds__(256) void moe_router(const float* __restrict__ x,
                                                  const float* __restrict__ rw,
                                                  float* __restrict__ combine) {
  const int lane = threadIdx.x & 31;
  const int wave = threadIdx.x >> 5;
  const int t = blockIdx.x * 8 + wave;

  float acc[EE];
#pragma unroll
  for (int e = 0; e < EE; ++e) acc[e] = 0.f;

  const float* xr = x + t * HH;
  for (int h = lane; h < HH; h += 32) {
    float xv = xr[h];
#pragma unroll
    for (int e = 0; e < EE; ++e) acc[e] += xv * rw[e * HH + h];
  }
#pragma unroll
  for (int e = 0; e < EE; ++e) {
    float v = acc[e];
#pragma unroll
    for (int off = 16; off > 0; off >>= 1) v += __shfl_xor(v, off, 32);
    acc[e] = v;
  }
  float mx = acc[0];
#pragma unroll
  for (int e = 1; e < EE; ++e) mx = fmaxf(mx, acc[e]);
  float p[EE];
  float sum = 0.f;
#pragma unroll
  for (int e = 0; e < EE; ++e) { p[e] = __expf(acc[e] - mx); sum += p[e]; }
  float inv = 1.f / sum;
#pragma unroll
  for (int e = 0; e < EE; ++e) p[e] *= inv;
  int i1 = 0;
#pragma unroll
  for (int e = 1; e < EE; ++e) if (p[e] > p[i1]) i1 = e;
  int i2 = (i1 == 0) ? 1 : 0;
#pragma unroll
  for (int e = 0; e < EE; ++e) if (e != i1 && p[e] > p[i2]) i2 = e;
  float denom = 1.f / (p[i1] + p[i2]);

  if (lane < EE) {
    float w = (lane == i1 || lane == i2) ? p[lane] * denom : 0.f;
    combine[t * EE + lane] = w;
  }
}

// ---------------------------------------------------------------- gate+up
// act[t][i] = silu(x @ gate_e)[t][i] * (x @ up_e)[t][i]
// Block tile 64(M) x 128(N); 8 waves as 2(M) x 4(N); wave tile 32x32.
// K loop over H in KC=32 chunks, double-buffered LDS staged via async DMA.
__global__ __launch_bounds__(256) void moe_gateup(const float* __restrict__ x,
                                                  const float* __restrict__ gw,
                                                  const float* __restrict__ uw,
                                                  float* __restrict__ act) {
  __shared__ float As[2][64][APAD];
  __shared__ float Bg[2][KC][BPAD];
  __shared__ float Bu[2][KC][BPAD];

  const int tid = threadIdx.x;
  const int lane = tid & 31;
  const int wave = tid >> 5;
  const int waveM = wave >> 2;  // 0..1
  const int waveN = wave & 3;   // 0..3
  const int tBase = blockIdx.y * 64;
  const int nBase0 = blockIdx.x * 128;
  const int l15 = lane & 15;
  const int klo = (lane >> 4) << 1;  // 0 or 2 (half-wave K split)

  // stage one KC chunk: A 64xKC (x rows), B KCx128 (gate & up rows).
  // 10 async b128 issues per thread per chunk.
  auto stage = [&](int buf, int h0) {
#pragma unroll
    for (int k = 0; k < 2; ++k) {
      int idx = tid + k * 256;
      int row = idx >> 3;        // 0..63
      int c4 = (idx & 7) << 2;   // 0,4,...,28
      async_b128(LDSOFF(&As[buf][row][c4]),
                 x + (tBase + row) * HH + h0 + c4);
    }
#pragma unroll
    for (int k = 0; k < 4; ++k) {
      int idx = tid + k * 256;
      int row = idx >> 5;        // 0..31
      int c4 = (idx & 31) << 2;  // 0,4,...,124
      async_b128(LDSOFF(&Bg[buf][row][c4]),
                 gw + (size_t)(h0 + row) * II + nBase0 + c4);
      async_b128(LDSOFF(&Bu[buf][row][c4]),
                 uw + (size_t)(h0 + row) * II + nBase0 + c4);
    }
  };

  const v8f zacc = {0.f, 0.f, 0.f, 0.f, 0.f, 0.f, 0.f, 0.f};
  v8f cg[2][2], cu[2][2];
#pragma unroll
  for (int mi = 0; mi < 2; ++mi)
#pragma unroll
    for (int ni = 0; ni < 2; ++ni) { cg[mi][ni] = zacc; cu[mi][ni] = zacc; }

  stage(0, 0);
  const int NCH = HH / KC;  // 32
  for (int hc = 0; hc < NCH; ++hc) {
    const int buf = hc & 1;
    if (hc + 1 < NCH) {
      stage(buf ^ 1, (hc + 1) * KC);
      WAIT_ASYNC(0xa);  // chunk hc's 10 issues retired (dones are in-order)
    } else {
      WAIT_ASYNC(0x0);
    }
    __syncthreads();

    const int ar = waveM * 32 + l15;
    const int c0 = waveN * 32 + l15;
#pragma unroll
    for (int k4 = 0; k4 < KC; k4 += 4) {
      const int kk = k4 + klo;
      v2f a0 = {As[buf][ar][kk], As[buf][ar][kk + 1]};
      v2f a1 = {As[buf][ar + 16][kk], As[buf][ar + 16][kk + 1]};
      v2f bg0 = {Bg[buf][kk][c0], Bg[buf][kk + 1][c0]};
      v2f bg1 = {Bg[buf][kk][c0 + 16], Bg[buf][kk + 1][c0 + 16]};
      v2f bu0 = {Bu[buf][kk][c0], Bu[buf][kk + 1][c0]};
      v2f bu1 = {Bu[buf][kk][c0 + 16], Bu[buf][kk + 1][c0 + 16]};

      cg[0][0] = wmma_f32(a0, bg0, cg[0][0]);
      cg[0][1] = wmma_f32(a0, bg1, cg[0][1]);
      cg[1][0] = wmma_f32(a1, bg0, cg[1][0]);
      cg[1][1] = wmma_f32(a1, bg1, cg[1][1]);
      cu[0][0] = wmma_f32(a0, bu0, cu[0][0]);
      cu[0][1] = wmma_f32(a0, bu1, cu[0][1]);
      cu[1][0] = wmma_f32(a1, bu0, cu[1][0]);
      cu[1][1] = wmma_f32(a1, bu1, cu[1][1]);
    }
    __syncthreads();  // protect buffer reuse by next iteration's stage()
  }

  // epilogue: act = silu(g) * u  (C layout: VGPR r -> row r + 8*(lane>=16))
#pragma unroll
  for (int mi = 0; mi < 2; ++mi) {
    const int rowBase = tBase + waveM * 32 + mi * 16 + ((lane >> 4) << 3);
#pragma unroll
    for (int ni = 0; ni < 2; ++ni) {
      const int col = nBase0 + waveN * 32 + ni * 16 + l15;
#pragma unroll
      for (int r = 0; r < 8; ++r) {
        float g = cg[mi][ni][r];
        float u = cu[mi][ni][r];
        float s = g / (1.f + __expf(-g));  // silu
        act[(rowBase + r) * II + col] = s * u;
      }
    }
  }
}

// ---------------------------------------------------------------- down proj
// out[t][h] += combine[t][e] * (act @ down_e)[t][h]
// Same tiling; K over I in KC chunks, async-LDS double buffered. Plain RMW on
// out (expert launches are stream-ordered, tiles disjoint) -> deterministic.
__global__ __launch_bounds__(256) void moe_down(const float* __restrict__ act,
                                                const float* __restrict__ dw,
                                                const float* __restrict__ combine,
                                                int e, float* __restrict__ out) {
  __shared__ float As[2][64][APAD];
  __shared__ float Bd[2][KC][BPAD];

  const int tid = threadIdx.x;
  const int lane = tid & 31;
  const int wave = tid >> 5;
  const int waveM = wave >> 2;
  const int waveN = wave & 3;
  const int tBase = blockIdx.y * 64;
  const int nBase0 = blockIdx.x * 128;
  const int l15 = lane & 15;
  const int klo = (lane >> 4) << 1;

  // 6 async b128 issues per thread per chunk.
  auto stage = [&](int buf, int k0) {
#pragma unroll
    for (int k = 0; k < 2; ++k) {
      int idx = tid + k * 256;
      int row = idx >> 3;
      int c4 = (idx & 7) << 2;
      async_b128(LDSOFF(&As[buf][row][c4]),
                 act + (tBase + row) * II + k0 + c4);
    }
#pragma unroll
    for (int k = 0; k < 4; ++k) {
      int idx = tid + k * 256;
      int row = idx >> 5;
      int c4 = (idx & 31) << 2;
      async_b128(LDSOFF(&Bd[buf][row][c4]),
                 dw + (size_t)(k0 + row) * HH + nBase0 + c4);
    }
  };

  const v8f zacc = {0.f, 0.f, 0.f, 0.f, 0.f, 0.f, 0.f, 0.f};
  v8f c[2][2];
#pragma unroll
  for (int mi = 0; mi < 2; ++mi)
#pragma unroll
    for (int ni = 0; ni < 2; ++ni) c[mi][ni] = zacc;

  stage(0, 0);
  const int NCH = II / KC;  // 44
  for (int kc = 0; kc < NCH; ++kc) {
    const int buf = kc & 1;
    if (kc + 1 < NCH) {
      stage(buf ^ 1, (kc + 1) * KC);
      WAIT_ASYNC(0x6);
    } else {
      WAIT_ASYNC(0x0);
    }
    __syncthreads();

    const int ar = waveM * 32 + l15;
    const int c0 = waveN * 32 + l15;
#pragma unroll
    for (int k4 = 0; k4 < KC; k4 += 4) {
      const int kk = k4 + klo;
      v2f a0 = {As[buf][ar][kk], As[buf][ar][kk + 1]};
      v2f a1 = {As[buf][ar + 16][kk], As[buf][ar + 16][kk + 1]};
      v2f b0 = {Bd[buf][kk][c0], Bd[buf][kk + 1][c0]};
      v2f b1 = {Bd[buf][kk][c0 + 16], Bd[buf][kk + 1][c0 + 16]};

      c[0][0] = wmma_f32(a0, b0, c[0][0]);
      c[0][1] = wmma_f32(a0, b1, c[0][1]);
      c[1][0] = wmma_f32(a1, b0, c[1][0]);
      c[1][1] = wmma_f32(a1, b1, c[1][1]);
    }
    __syncthreads();
  }

#pragma unroll
  for (int mi = 0; mi < 2; ++mi) {
    const int rowBase = tBase + waveM * 32 + mi * 16 + ((lane >> 4) << 3);
#pragma unroll
    for (int ni = 0; ni < 2; ++ni) {
      const int col = nBase0 + waveN * 32 + ni * 16 + l15;
#pragma unroll
      for (int r = 0; r < 8; ++r) {
        const int row = rowBase + r;
        float w = combine[row * EE + e];
        out[row * HH + col] += w * c[mi][ni][r];
      }
    }
  }
}

// ---------------------------------------------------------------- launch
extern "C" void kernel_launch(void* const* d_in, const int* in_sizes, int n_in,
                              void* d_out, int out_size, void* d_ws,
                              size_t ws_size, hipStream_t stream) {
  (void)in_sizes; (void)n_in; (void)out_size; (void)ws_size;
  const float* x    = (const float*)d_in[0];  // [T, H]
  const float* rw   = (const float*)d_in[1];  // [E, H]
  const float* gate = (const float*)d_in[2];  // [E, H, I]
  const float* up   = (const float*)d_in[3];  // [E, H, I]
  const float* down = (const float*)d_in[4];  // [E, I, H]
  float* out = (float*)d_out;                 // [T, H]

  float* combine = (float*)d_ws;              // [T, E]   32 KB
  float* act     = combine + (size_t)TT * EE; // [T, I]   5.8 MB

  moe_zero<<<(TT * HH) / (256 * 4), 256, 0, stream>>>((float4*)out);
  moe_router<<<TT / 8, 256, 0, stream>>>(x, rw, combine);

  for (int e = 0; e < EE; ++e) {
    const float* gwp = gate + (size_t)e * HH * II;
    const float* uwp = up + (size_t)e * HH * II;
    const float* dwp = down + (size_t)e * II * HH;
    moe_gateup<<<dim3(II / 128, TT / 64), 256, 0, stream>>>(x, gwp, uwp, act);
    moe_down<<<dim3(HH / 128, TT / 64), 256, 0, stream>>>(act, dwp, combine, e,
                                                          out);
  }
}